// Sampler_21672404975820
// MI455X (gfx1250) — compile-verified
//
#include <hip/hip_runtime.h>
#include <math.h>
#include <stdint.h>
#include <limits.h>

#define BB 128
#define VV 128000
#define KK 20
#define TPB 256
#define TILE 1024              // floats per tile = TPB * 4
#define SSEG 25                // segments per row
#define NT 5                   // tiles per segment (125/25)
#define SEGLEN (NT * TILE)     // 5120 elements per segment
#define NBLK (BB * SSEG)       // 3200 stage-1 blocks
#define SEPS 1e-5f

// workspace layout (dwords): m[NBLK] s[NBLK] gv[NBLK] gi[NBLK] rv[NBLK] ri[NBLK]
//                            tv[NBLK*KK] ti[NBLK*KK]
#define WS_M   0
#define WS_S   (NBLK)
#define WS_GV  (2 * NBLK)
#define WS_GI  (3 * NBLK)
#define WS_RV  (4 * NBLK)
#define WS_RI  (5 * NBLK)
#define WS_TV  (6 * NBLK)
#define WS_TI  (6 * NBLK + KK * NBLK)

__device__ __forceinline__ void lse_combine(float& m1, float& s1, float m2, float s2) {
  const float M  = fmaxf(m1, m2);
  const float e1 = (m1 == -INFINITY) ? 0.0f : __expf(m1 - M);
  const float e2 = (m2 == -INFINITY) ? 0.0f : __expf(m2 - M);
  s1 = s1 * e1 + s2 * e2;
  m1 = M;
}

// argmax with tie-break to smaller vocab index
__device__ __forceinline__ void amax_combine(float& v1, int& i1, float v2, int i2) {
  if (v2 > v1 || (v2 == v1 && i2 < i1)) { v1 = v2; i1 = i2; }
}

__global__ __launch_bounds__(TPB) void sampler_stage1(
    const float* __restrict__ logits,
    const float* __restrict__ temp,
    const float* __restrict__ u,
    float* __restrict__ wsf)
{
  const int seg = blockIdx.x;
  const int row = blockIdx.y;
  const int tid = threadIdx.x;
  const int bid = row * SSEG + seg;
  int* __restrict__ wsi = (int*)wsf;

  // 5 staging tile-pairs (all in flight at once) + per-thread candidate columns
  __shared__ float s_lg[NT][TILE];
  __shared__ float s_u [NT][TILE];
  __shared__ float s_tv[KK * TPB];
  __shared__ int   s_ti[KK * TPB];
  __shared__ float s_wm[8], s_ws[8], s_wgv[8], s_wrv[8], s_wv[8];
  __shared__ int   s_wgi[8], s_wri[8], s_wi[8], s_wo[8];

  const float* lgrow = logits + (size_t)row * VV + (size_t)seg * SEGLEN;
  const float* urow  = u      + (size_t)row * VV + (size_t)seg * SEGLEN;

  const float tr   = temp[row];
  const float invt = (tr < SEPS) ? 1.0f : (1.0f / tr);

  // ---- issue ALL tiles up front: 10 async b128 loads per wave (5KB/wave in flight) ----
#pragma unroll
  for (int t = 0; t < NT; ++t) {
    uint32_t goff = (uint32_t)((t * TILE + tid * 4) * 4);
    uint32_t l0 = (uint32_t)(uintptr_t)&s_lg[t][tid * 4];
    uint32_t l1 = (uint32_t)(uintptr_t)&s_u [t][tid * 4];
    asm volatile("global_load_async_to_lds_b128 %0, %1, %2 th:TH_LOAD_NT"
                 :: "v"(l0), "v"(goff), "s"(lgrow) : "memory");
    asm volatile("global_load_async_to_lds_b128 %0, %1, %2 th:TH_LOAD_NT"
                 :: "v"(l1), "v"(goff), "s"(urow)  : "memory");
  }
  asm volatile("s_wait_asynccnt 0x0" ::: "memory");
  __syncthreads();

  // ---- single pass over my 20 elements ----
  float m = -INFINITY, ssum = 0.0f;
  float gval = -INFINITY; int gidx = INT_MAX;
  float rval = -INFINITY; int ridx = INT_MAX;

#pragma unroll
  for (int t = 0; t < NT; ++t) {
    const float4 lg4 = *reinterpret_cast<const float4*>(&s_lg[t][tid * 4]);
    const float4 uu4 = *reinterpret_cast<const float4*>(&s_u [t][tid * 4]);
    const float lgv[4] = {lg4.x, lg4.y, lg4.z, lg4.w};
    const float uuv[4] = {uu4.x, uu4.y, uu4.z, uu4.w};
    const int base = seg * SEGLEN + t * TILE + tid * 4;
#pragma unroll
    for (int e = 0; e < 4; ++e) {
      const int   idx = base + e;
      const float sc  = lgv[e] * invt;
      if (sc > m) { ssum = ssum * __expf(m - sc) + 1.0f; m = sc; }
      else        { ssum += __expf(sc - m); }
      if (sc > gval) { gval = sc; gidx = idx; }
      const float r = sc - __logf(-__logf(uuv[e]));
      if (r > rval) { rval = r; ridx = idx; }
      const int slot = t * 4 + e;
      s_tv[slot * TPB + tid] = sc;      // every element is a candidate
      s_ti[slot * TPB + tid] = idx;
    }
  }

  // ---- wave32 butterfly reductions (barrier-free), then 8-leader merge ----
  {
    float bm = m, bs = ssum, bgv = gval, brv = rval;
    int bgi = gidx, bri = ridx;
#pragma unroll
    for (int d = 16; d >= 1; d >>= 1) {
      float m2  = __shfl_xor(bm, d, 32);
      float s2  = __shfl_xor(bs, d, 32);
      float gv2 = __shfl_xor(bgv, d, 32);
      int   gi2 = __shfl_xor(bgi, d, 32);
      float rv2 = __shfl_xor(brv, d, 32);
      int   ri2 = __shfl_xor(bri, d, 32);
      lse_combine(bm, bs, m2, s2);
      amax_combine(bgv, bgi, gv2, gi2);
      amax_combine(brv, bri, rv2, ri2);
    }
    if ((tid & 31) == 0) {
      const int w = tid >> 5;
      s_wm[w] = bm; s_ws[w] = bs;
      s_wgv[w] = bgv; s_wgi[w] = bgi;
      s_wrv[w] = brv; s_wri[w] = bri;
    }
  }
  __syncthreads();
  if (tid == 0) {
    float bm = s_wm[0], bs = s_ws[0], bgv = s_wgv[0], brv = s_wrv[0];
    int bgi = s_wgi[0], bri = s_wri[0];
    for (int w = 1; w < 8; ++w) {
      lse_combine(bm, bs, s_wm[w], s_ws[w]);
      amax_combine(bgv, bgi, s_wgv[w], s_wgi[w]);
      amax_combine(brv, bri, s_wrv[w], s_wri[w]);
    }
    wsf[WS_M  + bid] = bm;  wsf[WS_S  + bid] = bs;
    wsf[WS_GV + bid] = bgv; wsi[WS_GI + bid] = bgi;
    wsf[WS_RV + bid] = brv; wsi[WS_RI + bid] = bri;
  }

  // ---- insertion-sort my private column (desc by value, ties by asc index) ----
  for (int j = 1; j < KK; ++j) {
    const float v  = s_tv[j * TPB + tid];
    const int   ii = s_ti[j * TPB + tid];
    int p = j - 1;
    while (p >= 0) {
      const float vp = s_tv[p * TPB + tid];
      const int   ip = s_ti[p * TPB + tid];
      if (!((vp < v) || (vp == v && ip > ii))) break;
      s_tv[(p + 1) * TPB + tid] = vp;
      s_ti[(p + 1) * TPB + tid] = ip;
      --p;
    }
    s_tv[(p + 1) * TPB + tid] = v;
    s_ti[(p + 1) * TPB + tid] = ii;
  }

  // ---- 20 extraction rounds: O(1) head per thread + butterfly + 8-leader merge ----
  int ptr = 0;
  float hv = s_tv[tid];
  int   hi = s_ti[tid];
  for (int k = 0; k < KK; ++k) {
    float v = hv; int i = hi; int o = tid;
#pragma unroll
    for (int d = 16; d >= 1; d >>= 1) {
      float v2 = __shfl_xor(v, d, 32);
      int   i2 = __shfl_xor(i, d, 32);
      int   o2 = __shfl_xor(o, d, 32);
      if (v2 > v || (v2 == v && i2 < i)) { v = v2; i = i2; o = o2; }
    }
    if ((tid & 31) == 0) {
      const int w = tid >> 5;
      s_wv[w] = v; s_wi[w] = i; s_wo[w] = o;
    }
    __syncthreads();
    float bv = s_wv[0]; int bi = s_wi[0]; int bo = s_wo[0];
    for (int w = 1; w < 8; ++w) {
      const float v2 = s_wv[w]; const int i2 = s_wi[w];
      if (v2 > bv || (v2 == bv && i2 < bi)) { bv = v2; bi = i2; bo = s_wo[w]; }
    }
    if (tid == 0) {
      wsf[WS_TV + bid * KK + k] = bv;
      wsi[WS_TI + bid * KK + k] = bi;
    }
    if (tid == bo) {
      ++ptr;
      if (ptr < KK) { hv = s_tv[ptr * TPB + tid]; hi = s_ti[ptr * TPB + tid]; }
      else          { hv = -INFINITY; hi = INT_MAX; }
    }
    __syncthreads();
  }
}

__global__ __launch_bounds__(32) void sampler_stage2(
    const float* __restrict__ temp,
    const float* __restrict__ wsf,
    float* __restrict__ out)
{
  const int row  = blockIdx.x;
  const int lane = threadIdx.x;
  const int b    = row * SSEG + lane;
  const bool act = (lane < SSEG);
  const int* __restrict__ wsi = (const int*)wsf;

  float m  = act ? wsf[WS_M  + b] : -INFINITY;
  float s  = act ? wsf[WS_S  + b] : 0.0f;
  float gv = act ? wsf[WS_GV + b] : -INFINITY;
  int   gi = act ? wsi[WS_GI + b] : INT_MAX;
  float rv = act ? wsf[WS_RV + b] : -INFINITY;
  int   ri = act ? wsi[WS_RI + b] : INT_MAX;

#pragma unroll
  for (int d = 16; d >= 1; d >>= 1) {
    float m2  = __shfl_xor(m, d, 32);
    float s2  = __shfl_xor(s, d, 32);
    float gv2 = __shfl_xor(gv, d, 32);
    int   gi2 = __shfl_xor(gi, d, 32);
    float rv2 = __shfl_xor(rv, d, 32);
    int   ri2 = __shfl_xor(ri, d, 32);
    lse_combine(m, s, m2, s2);
    amax_combine(gv, gi, gv2, gi2);
    amax_combine(rv, ri, rv2, ri2);
  }
  const float lse = m + __logf(s);

  const float tr = temp[row];
  if (lane == 0) out[row] = (float)((tr < SEPS) ? gi : ri);

  // 25-way merge of sorted per-segment top-20 lists
  int p = 0;
  float hv = act ? wsf[WS_TV + b * KK] : -INFINITY;
  int   hi = act ? wsi[WS_TI + b * KK] : INT_MAX;
  for (int k = 0; k < KK; ++k) {
    float v = hv; int i = hi; int o = lane;
#pragma unroll
    for (int d = 16; d >= 1; d >>= 1) {
      float v2 = __shfl_xor(v, d, 32);
      int   i2 = __shfl_xor(i, d, 32);
      int   o2 = __shfl_xor(o, d, 32);
      if (v2 > v || (v2 == v && i2 < i)) { v = v2; i = i2; o = o2; }
    }
    if (lane == 0) {
      out[BB + row * KK + k]           = v - lse;
      out[BB + BB * KK + row * KK + k] = (float)i;
    }
    if (lane == o) {
      ++p;
      if (p < KK) { hv = wsf[WS_TV + b * KK + p]; hi = wsi[WS_TI + b * KK + p]; }
      else        { hv = -INFINITY; hi = INT_MAX; }
    }
  }
}

extern "C" void kernel_launch(void* const* d_in, const int* in_sizes, int n_in,
                              void* d_out, int out_size, void* d_ws, size_t ws_size,
                              hipStream_t stream) {
  const float* logits = (const float*)d_in[0];
  const float* temp   = (const float*)d_in[1];
  const float* u      = (const float*)d_in[2];
  float* out = (float*)d_out;
  float* wsf = (float*)d_ws;

  sampler_stage1<<<dim3(SSEG, BB), TPB, 0, stream>>>(logits, temp, u, wsf);
  sampler_stage2<<<BB, 32, 0, stream>>>(temp, wsf, out);
}